// GNNRetriever_22857815949345
// MI455X (gfx1250) — compile-verified
//
#include <hip/hip_runtime.h>

#define FD 64  // feature dim

typedef float v2f __attribute__((ext_vector_type(2)));
typedef float v8f __attribute__((ext_vector_type(8)));

// ---------------- utility kernels ----------------

__global__ void zero_f32(float* __restrict__ p, long n) {
  long i = (long)blockIdx.x * blockDim.x + threadIdx.x;
  if (i < n) p[i] = 0.0f;
}

__global__ void count_batch(const int* __restrict__ batch, float* __restrict__ cnt, int n) {
  int i = blockIdx.x * blockDim.x + threadIdx.x;
  if (i < n) unsafeAtomicAdd(&cnt[batch[i]], 1.0f);
}

// One thread per (edge, 4-float chunk): gather feat[src], scatter-add into msg[dst].
__global__ __launch_bounds__(256) void scatter_edges(
    const float* __restrict__ feat, const int* __restrict__ src,
    const int* __restrict__ dst, float* __restrict__ msg,
    float* __restrict__ deg /* null => skip degree */, int n_edges) {
  long idx = (long)blockIdx.x * blockDim.x + threadIdx.x;
  long total = (long)n_edges * (FD / 4);
  if (idx >= total) return;
  int e = (int)(idx >> 4);
  int c = (int)(idx & 15);
  int s = src[e];
  int d = dst[e];
  const float4 v = *(const float4*)(feat + (size_t)s * FD + c * 4);
  float* m = msg + (size_t)d * FD + c * 4;
  unsafeAtomicAdd(m + 0, v.x);
  unsafeAtomicAdd(m + 1, v.y);
  unsafeAtomicAdd(m + 2, v.z);
  unsafeAtomicAdd(m + 3, v.w);
  if (deg != nullptr && c == 0) unsafeAtomicAdd(deg + d, 1.0f);
}

__global__ void finalize_mean(float* __restrict__ msg, const float* __restrict__ deg, int n_nodes) {
  long i = (long)blockIdx.x * blockDim.x + threadIdx.x;
  long total = (long)n_nodes * FD;
  if (i >= total) return;
  int node = (int)(i >> 6);
  msg[i] = msg[i] / fmaxf(deg[node], 1.0f);
}

__global__ void final_div(const float* __restrict__ pool, const float* __restrict__ cnt,
                          float* __restrict__ out, int n) {
  int i = blockIdx.x * blockDim.x + threadIdx.x;
  if (i < n) out[i] = pool[i] / fmaxf(cnt[i >> 6], 1.0f);
}

// ---------------- fused SAGE linear layer via V_WMMA_F32_16X16X4_F32 ----------------
// out[m, :] = mean[m,:] @ Wl.T + bias + xin[m,:] @ Wr.T   (optionally ReLU)
// If outp==null: fused global mean-pool: atomicAdd(pool[batch[m]*64 + n], val).
// 256 threads = 8 waves; each wave owns a 16-row block; Wl/Wr staged in LDS (32.25 KB).
__global__ __launch_bounds__(256) void sage_gemm(
    const float* __restrict__ mean, const float* __restrict__ xin,
    const float* __restrict__ Wl, const float* __restrict__ Wr,
    const float* __restrict__ bias, float* __restrict__ outp,
    const int* __restrict__ batch, float* __restrict__ pool,
    int n_nodes, int do_relu) {
  __shared__ float sWl[FD * FD];
  __shared__ float sWr[FD * FD];
  __shared__ float sB[FD];

  for (int i = threadIdx.x; i < FD * FD; i += 256) {
    sWl[i] = Wl[i];
    sWr[i] = Wr[i];
  }
  if (threadIdx.x < FD) sB[threadIdx.x] = bias[threadIdx.x];
  __syncthreads();

  const int wave = threadIdx.x >> 5;
  const int lane = threadIdx.x & 31;
  const int rowbase = (blockIdx.x * 8 + wave) * 16;
  if (rowbase >= n_nodes) return;  // wave-uniform exit

  const int col  = lane & 15;   // N (and A-row) index within tile
  const int half = lane >> 4;   // K-pair selector for 32-bit fragments

  // Accumulators for the 4 N-tiles of the 16x64 output block; C preloaded with bias.
  v8f acc[4];
#pragma unroll
  for (int nt = 0; nt < 4; ++nt) {
    float b = sB[nt * 16 + col];
    acc[nt] = (v8f){b, b, b, b, b, b, b, b};
  }

  // A-fragment row for this lane (clamped so EXEC stays all-ones for WMMA).
  int rowA = rowbase + col;
  if (rowA > n_nodes - 1) rowA = n_nodes - 1;
  const float* mrow = mean + (size_t)rowA * FD;
  const float* xrow = xin + (size_t)rowA * FD;

  for (int kt = 0; kt < 16; ++kt) {
    const int kb = kt * 4 + half * 2;
    v2f aM = *(const v2f*)(mrow + kb);
    v2f aX = *(const v2f*)(xrow + kb);
#pragma unroll
    for (int nt = 0; nt < 4; ++nt) {
      // B[k][n] = W[n][k] (out = A @ W.T), W row-major in LDS.
      v2f bL = *(const v2f*)(sWl + (nt * 16 + col) * FD + kb);
      v2f bR = *(const v2f*)(sWr + (nt * 16 + col) * FD + kb);
      acc[nt] = __builtin_amdgcn_wmma_f32_16x16x4_f32(
          false, aM, false, bL, (short)0, acc[nt], false, false);
      acc[nt] = __builtin_amdgcn_wmma_f32_16x16x4_f32(
          false, aX, false, bR, (short)0, acc[nt], false, false);
    }
  }

  // D layout: VGPR v holds row rowbase + v + 8*half, column nt*16 + col.
#pragma unroll
  for (int nt = 0; nt < 4; ++nt) {
#pragma unroll
    for (int v = 0; v < 8; ++v) {
      int row = rowbase + v + 8 * half;
      if (row < n_nodes) {
        float val = acc[nt][v];
        if (do_relu) val = fmaxf(val, 0.0f);
        int n = nt * 16 + col;
        if (outp != nullptr) {
          outp[(size_t)row * FD + n] = val;
        } else {
          unsafeAtomicAdd(pool + (size_t)batch[row] * FD + n, val);
        }
      }
    }
  }
}

// ---------------- launch ----------------

extern "C" void kernel_launch(void* const* d_in, const int* in_sizes, int n_in,
                              void* d_out, int out_size, void* d_ws, size_t ws_size,
                              hipStream_t stream) {
  const float* x          = (const float*)d_in[0];
  const int*   edge_index = (const int*)d_in[1];
  // d_in[2] edge_attr and d_in[4] edge_emb feed a dead value in the reference.
  const int*   batch      = (const int*)d_in[3];
  const float* W1_l       = (const float*)d_in[5];
  const float* b1_l       = (const float*)d_in[6];
  const float* W1_r       = (const float*)d_in[7];
  const float* W2_l       = (const float*)d_in[8];
  const float* b2_l       = (const float*)d_in[9];
  const float* W2_r       = (const float*)d_in[10];
  float* out = (float*)d_out;

  const int N = in_sizes[0] / FD;   // 100000
  const int E = in_sizes[1] / 2;    // 1000000
  const int G = out_size / FD;      // 64

  // Workspace layout: [msg N*64][deg N][pool G*64][cnt G][h1 N*64]
  float* msg  = (float*)d_ws;
  float* deg  = msg + (size_t)N * FD;
  float* pool = deg + N;
  float* cnt  = pool + (size_t)G * FD;
  float* h1   = cnt + G;

  const int* src = edge_index;
  const int* dst = edge_index + E;

  const long nfeat = (long)N * FD;
  const long zn = nfeat + N + (long)G * FD + G;  // msg..cnt are contiguous
  const long sc = (long)E * (FD / 4);
  const int gemm_blocks = (N + 127) / 128;

  zero_f32<<<(int)((zn + 255) / 256), 256, 0, stream>>>(msg, zn);
  count_batch<<<(N + 255) / 256, 256, 0, stream>>>(batch, cnt, N);

  // Layer 1
  scatter_edges<<<(int)((sc + 255) / 256), 256, 0, stream>>>(x, src, dst, msg, deg, E);
  finalize_mean<<<(int)((nfeat + 255) / 256), 256, 0, stream>>>(msg, deg, N);
  sage_gemm<<<gemm_blocks, 256, 0, stream>>>(msg, x, W1_l, W1_r, b1_l, h1,
                                             nullptr, nullptr, N, /*relu=*/1);

  // Layer 2 (degree unchanged; reuse deg)
  zero_f32<<<(int)((nfeat + 255) / 256), 256, 0, stream>>>(msg, nfeat);
  scatter_edges<<<(int)((sc + 255) / 256), 256, 0, stream>>>(h1, src, dst, msg, nullptr, E);
  finalize_mean<<<(int)((nfeat + 255) / 256), 256, 0, stream>>>(msg, deg, N);
  sage_gemm<<<gemm_blocks, 256, 0, stream>>>(msg, h1, W2_l, W2_r, b2_l, nullptr,
                                             batch, pool, N, /*relu=*/0);

  final_div<<<(G * FD + 255) / 256, 256, 0, stream>>>(pool, cnt, out, G * FD);
}